// LRU_64587718197488
// MI455X (gfx1250) — compile-verified
//
#include <hip/hip_runtime.h>
#include <math.h>
#include <stdint.h>

// LRU scan: h[b,i,d] = lam[d]*h[b,i-1,d] + gam[d]*x[b,i,d]
// B=4, I=8192, D=1024, fp32.
// 3-phase blocked scan; phase 3 computes the intra-tile scan as
// D = A(16x16 lower-tri ones) x U via v_wmma_f32_16x16x4_f32 (4 K-slices),
// with exact per-channel diagonal rescaling lam^{-s} / lam^{t}.
// Tile staging uses GLOBAL_LOAD_ASYNC_TO_LDS_B128 (ASYNCcnt) when available,
// double-buffered; otherwise a software-prefetch ld/ds_store double buffer.

typedef float v2f __attribute__((ext_vector_type(2)));
typedef float v4f __attribute__((ext_vector_type(4)));
typedef float v8f __attribute__((ext_vector_type(8)));
typedef int   v4i __attribute__((vector_size(4 * sizeof(int))));

constexpr int Bc = 4;
constexpr int Ic = 8192;
constexpr int Dc = 1024;
constexpr int Lc = 128;           // chunk length (time steps)
constexpr int Cc = Ic / Lc;       // 64 chunks
constexpr int TW = 16;            // timesteps per WMMA tile
constexpr int CG = 128;           // channels per block in K3
constexpr int STR = CG + 8;       // LDS row stride (floats): +8 => halves hit disjoint banks

#if defined(__has_builtin)
#if __has_builtin(__builtin_amdgcn_global_load_async_to_lds_b128)
#define LRU_HAVE_ASYNC 1
#endif
#endif

#ifdef LRU_HAVE_ASYNC
typedef __attribute__((address_space(1))) v4i as1_v4i;
typedef __attribute__((address_space(3))) v4i as3_v4i;
#endif

// ---------------- K1: per-chunk local scan end values -------------------
// e[b,c,d] = sum_{s=0}^{L-1} lam^{L-1-s} * gam * x[b, c*L+s, d]
__global__ __launch_bounds__(256) void lru_k1_chunk_ends(
    const float* __restrict__ x, const float* __restrict__ nu_logs,
    float* __restrict__ e) {
  int gx    = blockIdx.x;               // B * C * (D/256) = 1024 blocks
  int dgrp  = gx % (Dc / 256);
  int chunk = (gx / (Dc / 256)) % Cc;
  int b     = gx / ((Dc / 256) * Cc);
  int d     = dgrp * 256 + threadIdx.x;

  float nv  = nu_logs[d];
  float lam = expf(-expf(nv));
  float gam = sqrtf(fmaxf(0.f, 1.f - lam * lam));

  const float* xp = x + ((size_t)b * Ic + (size_t)chunk * Lc) * Dc + d;
  float h = 0.f;
#pragma unroll 8
  for (int i = 0; i < Lc; ++i) {
    h = fmaf(lam, h, gam * xp[(size_t)i * Dc]);
  }
  e[((size_t)b * Cc + chunk) * Dc + d] = h;
}

// ---------------- K2: scan of chunk carries -----------------------------
// carry[b,0,d] = 0 ; carry[b,c,d] = lam^L * carry[b,c-1,d] + e[b,c-1,d]
__global__ __launch_bounds__(256) void lru_k2_carry_scan(
    const float* __restrict__ nu_logs, const float* __restrict__ e,
    float* __restrict__ carry) {
  int t = blockIdx.x * 256 + threadIdx.x;   // 0 .. B*D-1
  int d = t % Dc;
  int b = t / Dc;
  float nv   = nu_logs[d];
  float lamL = expf(-expf(nv) * (float)Lc);  // lam^L, exact in exponent space
  float c = 0.f;
  for (int k = 0; k < Cc; ++k) {
    size_t idx = ((size_t)b * Cc + k) * Dc + d;
    carry[idx] = c;
    c = fmaf(lamL, c, e[idx]);
  }
}

// ---------------- K3: WMMA-tiled intra-chunk scan + output --------------
__global__ __launch_bounds__(256) void lru_k3_wmma_scan(
    const float* __restrict__ x, const float* __restrict__ nu_logs,
    const float* __restrict__ carry_ws, float* __restrict__ out) {
  __shared__ alignas(16) float tin[2][TW * STR];   // double-buffered input tiles
  __shared__ alignas(16) float tout[TW * STR];     // output staging

  int gx    = blockIdx.x;                  // B * C * (D/CG) = 2048 blocks
  int group = gx % (Dc / CG);
  int chunk = (gx / (Dc / CG)) % Cc;
  int b     = gx / ((Dc / CG) * Cc);

  int tid    = threadIdx.x;
  int wave   = tid >> 5;
  int lane   = tid & 31;
  int lane16 = lane & 15;
  int half   = lane >> 4;
  int lc     = wave * 16 + lane16;         // local channel (this wave owns 16)
  int ch     = group * CG + lc;            // global channel

  // Per-channel parameters (per lane, channel = N = lane16 within wave group)
  float nv    = nu_logs[ch];
  float en    = expf(nv);
  float lam   = expf(-en);
  float gam   = sqrtf(fmaxf(0.f, 1.f - lam * lam));
  float linv  = expf(en);                  // 1/lam
  float linv2 = linv * linv, linv4 = linv2 * linv2;
  float lam2  = lam * lam, lam4 = lam2 * lam2, lam8 = lam4 * lam4;

  // A = 16x16 lower-triangular ones, split into 4 K-slices of 16x4.
  // ISA f32 A layout: lanes 0-15 M=0-15, lanes 16-31 M=0-15;
  // VGPR0 holds K=2*half+0, VGPR1 holds K=2*half+1 within a slice.
  v2f A[4];
#pragma unroll
  for (int j = 0; j < 4; ++j) {
    int k0 = 4 * j + 2 * half;
    A[j].x = (k0     <= lane16) ? 1.f : 0.f;
    A[j].y = (k0 + 1 <= lane16) ? 1.f : 0.f;
  }

  float hprev = carry_ws[((size_t)b * Cc + chunk) * Dc + ch];  // h at chunk start - 1
  size_t xbase = ((size_t)b * Ic + (size_t)chunk * Lc) * Dc + (size_t)group * CG;

  // Stage one 16x128 tile (16 rows x 512B, fully coalesced) into an LDS buffer.
  auto stage = [&](int td, float* buf) {
    int i0 = td * TW;
#pragma unroll
    for (int k2 = 0; k2 < 2; ++k2) {
      int idx = tid + 256 * k2;
      int t = idx >> 5, q = idx & 31;
      const float* gsrc = x + xbase + (size_t)(i0 + t) * Dc + q * 4;
      float* ldst = &buf[t * STR + q * 4];
#ifdef LRU_HAVE_ASYNC
      __builtin_amdgcn_global_load_async_to_lds_b128(
          (as1_v4i*)(uintptr_t)gsrc, (as3_v4i*)(uintptr_t)ldst, 0, 0);
#else
      *(v4f*)ldst = *(const v4f*)gsrc;
#endif
    }
  };

  stage(0, tin[0]);                          // prologue prefetch

  for (int td = 0; td < Lc / TW; ++td) {     // 8 tiles of 16 timesteps
    int cur = td & 1;
    if (td + 1 < Lc / TW) stage(td + 1, tin[cur ^ 1]);   // prefetch next tile

#ifdef LRU_HAVE_ASYNC
    // Wait for tile `td`'s 2 transfers; allow the next tile's 2 to stay in flight.
    if (td + 1 < Lc / TW) {
#if __has_builtin(__builtin_amdgcn_s_wait_asynccnt)
      __builtin_amdgcn_s_wait_asynccnt(2);
#else
      asm volatile("s_wait_asynccnt 2" ::: "memory");
#endif
    } else {
#if __has_builtin(__builtin_amdgcn_s_wait_asynccnt)
      __builtin_amdgcn_s_wait_asynccnt(0);
#else
      asm volatile("s_wait_asynccnt 0" ::: "memory");
#endif
    }
#endif
    __syncthreads();   // tile `cur` visible to all waves; tout free for rewrite

    // Build B (4x16 K-slices): U[s, c] = gam * lam^{-s} * x[s, c]
    const float* buf = tin[cur];
    v8f acc = {};
    float fj = gam * (half ? linv2 : 1.f);   // gam * lam^{-(2*half)}
#pragma unroll
    for (int j = 0; j < 4; ++j) {
      int s0 = 4 * j + 2 * half;
      v2f Bv;
      Bv.x = fj * buf[s0 * STR + lc];
      Bv.y = (fj * linv) * buf[(s0 + 1) * STR + lc];
      acc = __builtin_amdgcn_wmma_f32_16x16x4_f32(
          /*neg_a=*/false, A[j], /*neg_b=*/false, Bv,
          /*c_mod=*/(short)0, acc, /*reuse_a=*/false, /*reuse_b=*/false);
      fj *= linv4;                            // advance lam^{-4} per K-slice
    }

    // D layout: reg r holds M = r + 8*half, N = lane16.
    // h[t] = lam^t * (acc[t] + lam * hprev)
    float sc = half ? lam8 : 1.f;
    float lastv = 0.f;
#pragma unroll
    for (int r = 0; r < 8; ++r) {
      float h = sc * (acc[r] + lam * hprev);
      tout[(r + 8 * half) * STR + lc] = h;
      lastv = h;                              // reg7/half1 ends as H[15, c]
      sc *= lam;
    }
    // Broadcast H[15, c] (lives in lane 16+c, reg 7) to both halves.
    hprev = __shfl(lastv, 16 + lane16, 32);

    __syncthreads();   // tout complete
    // Cooperative LDS -> global, non-temporal (keep x resident in 192MB L2).
    int i0 = td * TW;
#pragma unroll
    for (int k2 = 0; k2 < 2; ++k2) {
      int idx = tid + 256 * k2;
      int t = idx >> 5, q = idx & 31;
      v4f v = *(const v4f*)&tout[t * STR + q * 4];
      __builtin_nontemporal_store(v, (v4f*)(out + xbase + (size_t)(i0 + t) * Dc + q * 4));
    }
    // next iteration's first barrier protects tout (write-after-read) and the
    // buffer being prefetched was last read in the previous iteration, before its tout barrier (reads complete
    // before each wave's barrier passage, DS ops are in-order per wave).
  }
}

extern "C" void kernel_launch(void* const* d_in, const int* in_sizes, int n_in,
                              void* d_out, int out_size, void* d_ws, size_t ws_size,
                              hipStream_t stream) {
  (void)in_sizes; (void)n_in; (void)out_size; (void)ws_size;
  const float* x  = (const float*)d_in[0];
  const float* nu = (const float*)d_in[1];
  float* out = (float*)d_out;

  float* e     = (float*)d_ws;                        // [B, C, D] = 1 MB
  float* carry = e + (size_t)Bc * Cc * Dc;            // [B, C, D] = 1 MB

  lru_k1_chunk_ends<<<Bc * Cc * (Dc / 256), 256, 0, stream>>>(x, nu, e);
  lru_k2_carry_scan<<<(Bc * Dc) / 256, 256, 0, stream>>>(nu, e, carry);
  lru_k3_wmma_scan<<<Bc * Cc * (Dc / CG), 256, 0, stream>>>(x, nu, carry, out);
}